// SparseDownsample_89678917140513
// MI455X (gfx1250) — compile-verified
//
#include <hip/hip_runtime.h>
#include <stdint.h>

// ---- problem constants (from reference) ----
#define C_CH        64
#define RES_OUT     128                        // RES/FACTOR
#define NBUCKETS    (2u * 128u * 128u * 128u)  // B * res^3 = 4,194,304
#define SCAN_THREADS 256
#define SCAN_ITEMS   16
#define SCAN_CHUNK   (SCAN_THREADS * SCAN_ITEMS)   // 4096
#define NB1          (NBUCKETS / SCAN_CHUNK)       // 1024 (exact)
#define PTS_PER_TILE 96                            // 96 pts * 64ch * 4B = 24KB per LDS buffer
#define TILES_PER_BLOCK 4                          // tiles per block (double-buffered TDM pipeline)

typedef unsigned int u32x4 __attribute__((ext_vector_type(4)));
typedef int          i32x4 __attribute__((ext_vector_type(4)));
typedef int          i32x8 __attribute__((ext_vector_type(8)));

// ---- TDM: 1-D contiguous tile of 4-byte elements, global -> LDS ----
__device__ __forceinline__ void tdm_load_1d(uint32_t lds_byte_off,
                                            const void* gptr,
                                            uint32_t nelem /* 4B elems, < 65536 */) {
  uint64_t ga = (uint64_t)(uintptr_t)gptr;
  u32x4 g0;
  g0[0] = 1u;                                   // [1:0] count=1 (valid user descriptor)
  g0[1] = lds_byte_off;                         // [63:32] lds_addr
  g0[2] = (uint32_t)ga;                         // [95:64] global_addr lo
  g0[3] = (uint32_t)((ga >> 32) & 0x01FFFFFFu)  // [120:96] global_addr hi (57-bit)
        | (2u << 30);                           // [127:126] type=2 ("image")
  i32x8 g1;
  g1[0] = (int)(2u << 16);                      // data_size=2 (4 bytes); mask/flags 0
  g1[1] = (int)((nelem & 0xFFFFu) << 16);       // tensor_dim0[15:0] at bits[63:48]
  g1[2] = (int)(((nelem >> 16) & 0xFFFFu)       // tensor_dim0[31:16]
        | (1u << 16));                          // tensor_dim1 = 1
  g1[3] = (int)((nelem & 0xFFFFu) << 16);       // tile_dim0 at bits[127:112]
  g1[4] = 1;                                    // tile_dim1 = 1, tile_dim2 = 0
  g1[5] = (int)nelem;                           // tensor_dim0_stride lo (single line)
  g1[6] = 0;
  g1[7] = 0;
  i32x4 z4 = {0, 0, 0, 0};                      // groups 2/3 unused (<=2D tile)
  i32x8 z8 = {0, 0, 0, 0, 0, 0, 0, 0};
  __builtin_amdgcn_tensor_load_to_lds(g0, g1, z4, z4, z8, 0);
}

__device__ __forceinline__ int bucket_code(int4 c) {
  // ((b*res + x/2)*res + y/2)*res + z/2
  return ((c.x * RES_OUT + (c.y >> 1)) * RES_OUT + (c.z >> 1)) * RES_OUT + (c.w >> 1);
}

// ---- pass 1: per-bucket histogram ----
__global__ void sd_count(const int* __restrict__ coords,
                         unsigned int* __restrict__ counts, int n) {
  int i = blockIdx.x * blockDim.x + threadIdx.x;
  if (i < n) {
    int4 c = ((const int4*)coords)[i];
    atomicAdd(&counts[bucket_code(c)], 1u);
  }
}

// ---- pass 2a: per-block exclusive scan of occupancy (4096 buckets / block) ----
__global__ void sd_scan1(const unsigned int* __restrict__ counts,
                         unsigned int* __restrict__ rankp,
                         unsigned int* __restrict__ bsums) {
  __shared__ unsigned int s[SCAN_THREADS];
  const int tid = threadIdx.x;
  const int base = blockIdx.x * SCAN_CHUNK + tid * SCAN_ITEMS;
  uint4 cin[4];
  const uint4* cp = (const uint4*)(counts + base);
#pragma unroll
  for (int q = 0; q < 4; ++q) cin[q] = cp[q];
  unsigned int occ[SCAN_ITEMS];
  unsigned int run = 0;
#pragma unroll
  for (int k = 0; k < SCAN_ITEMS; ++k) {
    unsigned int c = ((const unsigned int*)cin)[k];
    occ[k] = run;
    run += (c != 0u) ? 1u : 0u;
  }
  s[tid] = run;
  __syncthreads();
  for (int off = 1; off < SCAN_THREADS; off <<= 1) {  // Hillis-Steele inclusive
    unsigned int v = (tid >= off) ? s[tid - off] : 0u;
    __syncthreads();
    s[tid] += v;
    __syncthreads();
  }
  const unsigned int excl = s[tid] - run;
  uint4* rp = (uint4*)(rankp + base);
#pragma unroll
  for (int q = 0; q < 4; ++q) {
    uint4 o;
    o.x = excl + occ[4 * q + 0];
    o.y = excl + occ[4 * q + 1];
    o.z = excl + occ[4 * q + 2];
    o.w = excl + occ[4 * q + 3];
    rp[q] = o;
  }
  if (tid == SCAN_THREADS - 1) bsums[blockIdx.x] = s[tid];
}

// ---- pass 2b: scan the 1024 block sums (single block); bsums[NB1] = n_unique ----
__global__ void sd_scan2(unsigned int* __restrict__ bsums) {
  __shared__ unsigned int s[SCAN_THREADS];
  const int tid = threadIdx.x;
  const int base = tid * 4;
  unsigned int v[4];
  unsigned int run = 0;
#pragma unroll
  for (int k = 0; k < 4; ++k) {
    unsigned int x = bsums[base + k];
    v[k] = run;
    run += x;
  }
  s[tid] = run;
  __syncthreads();
  for (int off = 1; off < SCAN_THREADS; off <<= 1) {
    unsigned int t = (tid >= off) ? s[tid - off] : 0u;
    __syncthreads();
    s[tid] += t;
    __syncthreads();
  }
  const unsigned int excl = s[tid] - run;
#pragma unroll
  for (int k = 0; k < 4; ++k) bsums[base + k] = excl + v[k];
  if (tid == SCAN_THREADS - 1) bsums[NB1] = s[tid];  // total unique rows
}

// ---- pass 3: finalize ranks; emit sorted coords + per-row counts ----
__global__ void sd_buckets(const unsigned int* __restrict__ counts,
                           unsigned int* __restrict__ rankp,
                           const unsigned int* __restrict__ bsums,
                           float* __restrict__ outC,
                           unsigned int* __restrict__ rowCount) {
  const unsigned int b = blockIdx.x * blockDim.x + threadIdx.x;  // < NBUCKETS
  const unsigned int full = rankp[b] + bsums[b / SCAN_CHUNK];
  rankp[b] = full;
  const unsigned int cnt = counts[b];
  if (cnt) {
    rowCount[full] = cnt;
    float4 o;
    o.x = (float)(b >> 21);          // batch
    o.y = (float)((b >> 14) & 127);  // x
    o.z = (float)((b >> 7) & 127);   // y
    o.w = (float)(b & 127);          // z
    ((float4*)outC)[full] = o;
  }
}

// ---- pass 4: double-buffered TDM-staged scatter-add of features ----
__global__ void sd_scatter(const float* __restrict__ feats,
                           const int* __restrict__ coords,
                           const unsigned int* __restrict__ rankp,
                           float* __restrict__ outF, int n) {
  __shared__ float s_tile[2][PTS_PER_TILE * C_CH];   // 2 x 24KB
  __shared__ unsigned int s_row[PTS_PER_TILE];
  const int tid = threadIdx.x;
  const bool w0 = (tid < 32);                        // wave 0 drives the TDM
  const int total_tiles = (n + PTS_PER_TILE - 1) / PTS_PER_TILE;
  const int tile0 = blockIdx.x * TILES_PER_BLOCK;
  int nt = total_tiles - tile0;
  if (nt <= 0) return;
  if (nt > TILES_PER_BLOCK) nt = TILES_PER_BLOCK;

  // prologue: kick off DMA for tile 0
  if (w0) {
    int p0 = tile0 * PTS_PER_TILE;
    int m = n - p0; if (m > PTS_PER_TILE) m = PTS_PER_TILE;
    tdm_load_1d((uint32_t)(uintptr_t)&s_tile[0][0],
                feats + (size_t)p0 * C_CH, (uint32_t)(m * C_CH));
  }

  for (int t = 0; t < nt; ++t) {
    const int p0 = (tile0 + t) * PTS_PER_TILE;
    int npts = n - p0;
    if (npts > PTS_PER_TILE) npts = PTS_PER_TILE;

    // issue next tile's DMA into the other buffer before touching this one
    if (w0 && (t + 1 < nt)) {
      int p1 = p0 + PTS_PER_TILE;
      int m = n - p1; if (m > PTS_PER_TILE) m = PTS_PER_TILE;
      tdm_load_1d((uint32_t)(uintptr_t)&s_tile[(t + 1) & 1][0],
                  feats + (size_t)p1 * C_CH, (uint32_t)(m * C_CH));
    }
    // overlap: compute output row per point while DMA(s) in flight
    if (tid < npts) {
      int4 c = ((const int4*)coords)[p0 + tid];
      s_row[tid] = rankp[bucket_code(c)];
    }
    // TDM completes in order per wave: cnt<=1 means tile t has landed
    if (w0) {
      if (t + 1 < nt) __builtin_amdgcn_s_wait_tensorcnt(1);
      else            __builtin_amdgcn_s_wait_tensorcnt(0);
    }
    __syncthreads();

    const float* buf = &s_tile[t & 1][0];
    const int total = npts * C_CH;
    for (int i = tid; i < total; i += blockDim.x) {
      const unsigned int row = s_row[i >> 6];
      atomicAdd(&outF[(size_t)row * C_CH + (i & 63)], buf[i]);  // L2-resident
    }
    __syncthreads();  // buffer + s_row reuse fence
  }
}

// ---- pass 5: divide by counts; pad invalid coord rows with -1 ----
__global__ void sd_finalize(float* __restrict__ outF, float* __restrict__ outC,
                            const unsigned int* __restrict__ rowCount,
                            const unsigned int* __restrict__ nuniq_ptr, int n) {
  const unsigned int nu = *nuniq_ptr;
  const int i = blockIdx.x * blockDim.x + threadIdx.x;
  if (i < n * C_CH) {
    const int row = i >> 6;
    if ((unsigned int)row < nu) outF[i] = outF[i] / (float)rowCount[row];
    // rows >= nu stay 0 from the memset
  }
  if (i < n * 4) {
    if ((unsigned int)(i >> 2) >= nu) outC[i] = -1.0f;
  }
}

extern "C" void kernel_launch(void* const* d_in, const int* in_sizes, int n_in,
                              void* d_out, int out_size, void* d_ws, size_t ws_size,
                              hipStream_t stream) {
  const float* feats  = (const float*)d_in[0];
  const int*   coords = (const int*)d_in[1];
  const int n = in_sizes[1] / 4;  // N points

  unsigned int* counts   = (unsigned int*)d_ws;          // NBUCKETS
  unsigned int* rankp    = counts + NBUCKETS;            // NBUCKETS
  unsigned int* bsums    = rankp + NBUCKETS;             // NB1 + 1 (+pad)
  unsigned int* rowCount = bsums + (NB1 + 8);            // N
  float* outF = (float*)d_out;                           // [N, 64]
  float* outC = outF + (size_t)n * C_CH;                 // [N, 4]

  (void)hipMemsetAsync(counts, 0, (size_t)NBUCKETS * sizeof(unsigned int), stream);
  (void)hipMemsetAsync(d_out, 0, (size_t)out_size * sizeof(float), stream);

  const int total_tiles   = (n + PTS_PER_TILE - 1) / PTS_PER_TILE;
  const int scatter_blocks = (total_tiles + TILES_PER_BLOCK - 1) / TILES_PER_BLOCK;

  sd_count   <<<(n + 255) / 256, 256, 0, stream>>>(coords, counts, n);
  sd_scan1   <<<NB1, SCAN_THREADS, 0, stream>>>(counts, rankp, bsums);
  sd_scan2   <<<1, SCAN_THREADS, 0, stream>>>(bsums);
  sd_buckets <<<NBUCKETS / 256, 256, 0, stream>>>(counts, rankp, bsums, outC, rowCount);
  sd_scatter <<<scatter_blocks, 256, 0, stream>>>(feats, coords, rankp, outF, n);
  sd_finalize<<<(unsigned int)(((size_t)n * C_CH + 255) / 256), 256, 0, stream>>>(
      outF, outC, rowCount, bsums + NB1, n);
}